// VGAE_11158325035212
// MI455X (gfx1250) — compile-verified
//
#include <hip/hip_runtime.h>
#include <hip/hip_bf16.h>
#include <math.h>

// ---------------------------------------------------------------------------
// VGAE forward for MI455X (gfx1250, wave32, WMMA, TDM).
// Dominant stage: A_hat = sigmoid(Z @ Z^T) -> 400 MB output, ~17us HBM floor.
// All dense GEMMs use V_WMMA_F32_16X16X4_F32 (exact f32, K-step = 4).
// Z@Z^T is LDS-macro-tiled (2x4 tiles / 8 waves per block) with the Z rows
// staged via TENSOR_LOAD_TO_LDS (pad fields give a bank-conflict-free stride).
// ---------------------------------------------------------------------------

typedef __attribute__((ext_vector_type(2))) float v2f;
typedef __attribute__((ext_vector_type(8))) float v8f;
typedef unsigned int u32x4 __attribute__((ext_vector_type(4)));
typedef int          i32x4 __attribute__((ext_vector_type(4)));
typedef int          i32x8 __attribute__((ext_vector_type(8)));

#define GN    10000
#define GE    320000
#define GFIN  256
#define GHID  512
#define GLAT  64
#define LDS_STRIDE 68   // 64 + 4 pad floats -> 4-bank step per row, conflict-free

// ---------------------------------------------------------------------------
// row_start[i] = lower_bound(edge_rows, i)   (edge_rows is sorted)
// ---------------------------------------------------------------------------
__global__ void build_row_start(const int* __restrict__ rows,
                                int* __restrict__ row_start,
                                int E, int Nn) {
  int i = blockIdx.x * blockDim.x + threadIdx.x;
  if (i > Nn) return;
  int lo = 0, hi = E;
  while (lo < hi) {
    int mid = (lo + hi) >> 1;
    if (rows[mid] < i) lo = mid + 1; else hi = mid;
  }
  row_start[i] = lo;
}

// ---------------------------------------------------------------------------
// Generic f32 WMMA GEMM: C[M,Nn] = A[M,K] @ B[K,Nn]  (row-major, M%16==0,
// K%4==0, Nn%16==0). One wave computes one 16x16 tile via 16x16x4 f32 WMMA.
// ISA 7.12.2 layouts (wave32):
//   A 16x4 : VGPR v, lane L -> A[m0 + L%16][k0 + v + (L>=16 ? 2 : 0)]
//   B 4x16 : VGPR v, lane L -> B[k0 + v + (L>=16 ? 2 : 0)][n0 + L%16]
//   C/D    : VGPR r, lane L -> C[m0 + r + (L>=16 ? 8 : 0)][n0 + L%16]
// ---------------------------------------------------------------------------
__global__ void wmma_gemm_f32(const float* __restrict__ A,
                              const float* __restrict__ B,
                              float* __restrict__ C,
                              int M, int K, int Nn) {
  const int wave = (int)((blockIdx.x * blockDim.x + threadIdx.x) >> 5);
  const int lane = (int)(threadIdx.x & 31);
  const int tn = Nn >> 4;
  const int ntiles = (M >> 4) * tn;
  if (wave >= ntiles) return;                 // wave-uniform: EXEC stays all-1s
  const int ti = wave / tn;
  const int tj = wave - ti * tn;

  const int lm    = lane & 15;
  const int khalf = (lane < 16) ? 0 : 2;

  const float* __restrict__ arow = A + (size_t)((ti << 4) + lm) * K + khalf;
  const float* __restrict__ bcol = B + (size_t)khalf * Nn + ((tj << 4) + lm);

  v8f acc = {};
#pragma unroll 4
  for (int k0 = 0; k0 < K; k0 += 4) {
    v2f a = *(const v2f*)(arow + k0);         // 8B-aligned
    v2f b;
    b.x = bcol[(size_t)k0 * Nn];
    b.y = bcol[(size_t)(k0 + 1) * Nn];
    acc = __builtin_amdgcn_wmma_f32_16x16x4_f32(false, a, false, b,
                                                (short)0, acc, false, false);
  }

  const int row0 = (ti << 4) + ((lane < 16) ? 0 : 8);
  const int col  = (tj << 4) + lm;
#pragma unroll
  for (int r = 0; r < 8; ++r)
    C[(size_t)(row0 + r) * Nn + col] = acc[r];
}

// ---------------------------------------------------------------------------
// SpMM (CSR-by-binary-search): Hout[i,f] = sum_e vals[e]*Hin[cols[e],f],
// optional ReLU. Hin is L2-resident (<=20.5 MB of 192 MB), so the per-edge
// row gathers hit L2, not HBM.
// ---------------------------------------------------------------------------
__global__ void spmm_f32(const int* __restrict__ row_start,
                         const int* __restrict__ cols,
                         const float* __restrict__ vals,
                         const float* __restrict__ Hin,
                         float* __restrict__ Hout,
                         int F, int do_relu) {
  const int i = (int)blockIdx.x;
  const int f = (int)(blockIdx.y * blockDim.x + threadIdx.x);
  if (f >= F) return;
  const int e0 = row_start[i];
  const int e1 = row_start[i + 1];
  float acc = 0.0f;
  for (int e = e0; e < e1; ++e) {
    const int   c = cols[e];                  // uniform -> scalar load
    const float v = vals[e];                  // uniform -> scalar load
    acc = fmaf(v, Hin[(size_t)c * F + f], acc);
  }
  if (do_relu) acc = fmaxf(acc, 0.0f);
  Hout[(size_t)i * F + f] = acc;
}

// ---------------------------------------------------------------------------
// Z = mu + eps * exp(logsigma)
// ---------------------------------------------------------------------------
__global__ void reparam_f32(const float* __restrict__ mu,
                            const float* __restrict__ ls,
                            const float* __restrict__ eps,
                            float* __restrict__ Z, int n) {
  int t = (int)(blockIdx.x * blockDim.x + threadIdx.x);
  if (t < n) Z[t] = fmaf(eps[t], __expf(ls[t]), mu[t]);
}

// ---------------------------------------------------------------------------
// TDM: stage `ndw` contiguous DWORDs of Z rows into LDS, inserting 4 pad
// DWORDs after every 64 DWORDs (-> LDS_STRIDE=68 floats per 64-float row).
// D# group0: count=1, lds_addr, 57b global_addr, type=2.
// D# group1: data_size=4B(2), pad_enable, pad_interval=5(64dw), pad_amount=3
// (4dw); 1-D tile: tile_dim0=ndw, tensor_dim0=ndw, tensor_dim1=tile_dim1=1.
// ---------------------------------------------------------------------------
#if __has_builtin(__builtin_amdgcn_tensor_load_to_lds)
#define USE_TDM 1
__device__ __forceinline__ void tdm_load_rows_padded(unsigned lds_byte_addr,
                                                     const float* gsrc,
                                                     unsigned ndw) {
  unsigned long long ga = (unsigned long long)gsrc;
  u32x4 g0;
  g0[0] = 1u;                                            // count=1
  g0[1] = lds_byte_addr;                                 // lds_addr
  g0[2] = (unsigned)(ga & 0xFFFFFFFFull);                // global_addr[31:0]
  g0[3] = (unsigned)((ga >> 32) & 0x01FFFFFFull)         // global_addr[56:32]
        | (2u << 30);                                    // type=2 ("image")
  i32x8 g1;
  g1[0] = (int)((2u << 16) | (1u << 20) | (5u << 22) | (3u << 25));
  g1[1] = (int)((ndw & 0xFFFFu) << 16);                  // tensor_dim0[15:0]
  g1[2] = (int)(((ndw >> 16) & 0xFFFFu) | (1u << 16));   // dim0[31:16]|dim1=1
  g1[3] = (int)((ndw & 0xFFFFu) << 16);                  // tile_dim0 = ndw
  g1[4] = 1;                                             // tile_dim1 = 1
  g1[5] = (int)ndw;                                      // dim0_stride
  g1[6] = 0;
  g1[7] = 0;
  i32x4 z4 = {0, 0, 0, 0};
#if __clang_major__ >= 23
  i32x8 z8 = {0, 0, 0, 0, 0, 0, 0, 0};
  __builtin_amdgcn_tensor_load_to_lds(g0, g1, z4, z4, z8, 0);
#else
  __builtin_amdgcn_tensor_load_to_lds(g0, g1, z4, z4, 0);
#endif
}
#else
#define USE_TDM 0
#endif

__device__ __forceinline__ unsigned lds_addr_of(const void* p) {
  // LDS aperture base has zero low 32 bits; flat-addr[31:0] is the LDS offset.
  return (unsigned)(unsigned long long)p;
}

// ---------------------------------------------------------------------------
// A_hat = sigmoid(Z @ Z^T), Z:[Nn,64]. Block = 256 thr = 8 waves computing a
// 2x4 grid of 16x16 tiles (32 rows x 64 cols out). Z i-rows (32) and j-rows
// (64) are staged once into LDS (24 KB + pad), cutting L2 fragment reads
// ~2.7x so the 400 MB store stream stays the bottleneck. Fragments then come
// from ds_load_b64 at a conflict-free padded stride.
// ---------------------------------------------------------------------------
__global__ void __launch_bounds__(256)
zzt_sigmoid_lds(const float* __restrict__ Z, float* __restrict__ out, int Nn) {
  __shared__ float lds_i[32 * LDS_STRIDE];
  __shared__ float lds_j[64 * LDS_STRIDE];

  const int tid  = (int)threadIdx.x;
  const int lane = tid & 31;
  const int wv   = tid >> 5;                  // 0..7
  const int i_row0 = (int)blockIdx.y * 32;
  const int j_row0 = (int)blockIdx.x * 64;

#if USE_TDM
  if (wv == 0) {                              // wave-uniform branch
    tdm_load_rows_padded(lds_addr_of(lds_i), Z + (size_t)i_row0 * GLAT,
                         32u * GLAT);
    tdm_load_rows_padded(lds_addr_of(lds_j), Z + (size_t)j_row0 * GLAT,
                         64u * GLAT);
    __builtin_amdgcn_s_wait_tensorcnt(0);
  }
#else
  // Cooperative fallback: 4 threads per row, float4 chunks, padded stride.
  for (int rr = tid >> 2; rr < 96; rr += 64) {
    const int isI = rr < 32;
    const int gr  = isI ? (i_row0 + rr) : (j_row0 + (rr - 32));
    const float4* src = (const float4*)(Z + (size_t)gr * GLAT);
    float4* dst = (float4*)(isI ? (lds_i + rr * LDS_STRIDE)
                                : (lds_j + (rr - 32) * LDS_STRIDE));
#pragma unroll
    for (int c = (tid & 3); c < 16; c += 4) dst[c] = src[c];
  }
#endif
  __syncthreads();

  const int ti = (int)blockIdx.y * 2 + (wv >> 2);
  const int tj = (int)blockIdx.x * 4 + (wv & 3);
  if (ti < (GN >> 4) && tj < (GN >> 4)) {     // wave-uniform guard
    const int lm    = lane & 15;
    const int khalf = (lane < 16) ? 0 : 2;
    const float* za = lds_i + ((wv >> 2) * 16 + lm) * LDS_STRIDE + khalf;
    const float* zb = lds_j + ((wv & 3)  * 16 + lm) * LDS_STRIDE + khalf;

    v8f acc = {};
#pragma unroll
    for (int k0 = 0; k0 < GLAT; k0 += 4) {
      v2f a = *(const v2f*)(za + k0);         // ds_load_b64, 8B-aligned
      v2f b = *(const v2f*)(zb + k0);
      acc = __builtin_amdgcn_wmma_f32_16x16x4_f32(false, a, false, b,
                                                  (short)0, acc, false, false);
    }

    const int row0 = (ti << 4) + ((lane < 16) ? 0 : 8);
    const int col  = (tj << 4) + lm;
#pragma unroll
    for (int r = 0; r < 8; ++r) {
      float s = 1.0f / (1.0f + __expf(-acc[r]));
      out[(size_t)(row0 + r) * Nn + col] = s; // 16-wide coalesced row stores
    }
  }
}

// ---------------------------------------------------------------------------
// Host-side orchestration
// ---------------------------------------------------------------------------
extern "C" void kernel_launch(void* const* d_in, const int* in_sizes, int n_in,
                              void* d_out, int out_size, void* d_ws, size_t ws_size,
                              hipStream_t stream) {
  (void)in_sizes; (void)n_in; (void)out_size; (void)ws_size;

  const float* X    = (const float*)d_in[0];   // [N, F_IN]
  const float* vals = (const float*)d_in[1];   // [E]
  const float* W1   = (const float*)d_in[2];   // [F_IN, HID]
  const float* Wmu  = (const float*)d_in[3];   // [HID, LAT]
  const float* Wls  = (const float*)d_in[4];   // [HID, LAT]
  const float* eps  = (const float*)d_in[5];   // [N, LAT]
  const int*   rows = (const int*)d_in[6];     // [E] sorted
  const int*   cols = (const int*)d_in[7];     // [E]

  float* out = (float*)d_out;
  float* out_Ahat = out;                                   // N*N
  float* out_mu   = out + (size_t)GN * GN;                 // N*LAT
  float* out_ls   = out + (size_t)GN * GN + (size_t)GN * GLAT;

  // Workspace carve-up (bytes)
  char* ws = (char*)d_ws;
  float* xw1    = (float*)(ws);                                       // N*HID
  float* h      = (float*)(ws + (size_t)GN * GHID * 4);               // N*HID
  float* hwmu   = (float*)(ws + (size_t)GN * GHID * 8);               // N*LAT
  float* hwls   = (float*)(ws + (size_t)GN * GHID * 8 + (size_t)GN * GLAT * 4);
  float* Zbuf   = (float*)(ws + (size_t)GN * GHID * 8 + (size_t)GN * GLAT * 8);
  int*   rstart = (int*)  (ws + (size_t)GN * GHID * 8 + (size_t)GN * GLAT * 12);

  // 1) CSR row pointers from sorted COO rows
  build_row_start<<<(GN + 1 + 255) / 256, 256, 0, stream>>>(rows, rstart, GE, GN);

  // 2) XW1 = X @ W1  (tiles: 625*32 = 20000; 8 waves/block)
  wmma_gemm_f32<<<20000 / 8, 256, 0, stream>>>(X, W1, xw1, GN, GFIN, GHID);

  // 3) h = relu(A_tilde @ XW1)
  spmm_f32<<<dim3(GN, GHID / 256), 256, 0, stream>>>(rstart, cols, vals, xw1, h,
                                                     GHID, 1);

  // 4) hWmu = h @ Wmu, hWls = h @ Wls  (tiles: 625*4 = 2500)
  wmma_gemm_f32<<<(2500 + 7) / 8, 256, 0, stream>>>(h, Wmu, hwmu, GN, GHID, GLAT);
  wmma_gemm_f32<<<(2500 + 7) / 8, 256, 0, stream>>>(h, Wls, hwls, GN, GHID, GLAT);

  // 5) mu / logsigma SpMMs straight into d_out
  spmm_f32<<<dim3(GN, 1), GLAT, 0, stream>>>(rstart, cols, vals, hwmu, out_mu,
                                             GLAT, 0);
  spmm_f32<<<dim3(GN, 1), GLAT, 0, stream>>>(rstart, cols, vals, hwls, out_ls,
                                             GLAT, 0);

  // 6) Z = mu + eps * exp(logsigma)
  reparam_f32<<<(GN * GLAT + 255) / 256, 256, 0, stream>>>(out_mu, out_ls, eps,
                                                           Zbuf, GN * GLAT);

  // 7) A_hat = sigmoid(Z @ Z^T): 157 x 313 macro-blocks of 2x4 tiles
  zzt_sigmoid_lds<<<dim3((625 + 3) / 4, (625 + 1) / 2), 256, 0, stream>>>(
      Zbuf, out_Ahat, GN);
}